// PairwisePredictionHead_32083405701179
// MI455X (gfx1250) — compile-verified
//
#include <hip/hip_runtime.h>
#include <math.h>

// ---------------------------------------------------------------------------
// PairwisePredictionHead for MI455X (gfx1250, wave32, WMMA f32 16x16x4)
//
// Factorization: with W1p = W1[:, :64], W1d = W1[:, 64:],
//   h1[i,j,:] = W1p.(q_j * k_i)  +  Aq[j] - Ak[i] + b1,
//   Aq = q @ W1d^T, Ak = k @ W1d^T  (precomputed, fused into downproj)
// -> halves GEMM1 K-depth (128 -> 64): total matrix FLOPs 29 GF -> 19.3 GF.
// ---------------------------------------------------------------------------

typedef float v2f __attribute__((ext_vector_type(2)));
typedef float v8f __attribute__((ext_vector_type(8)));

#define L_SEQ   768
#define D_IN    1024
#define DP      128
#define NB      64
#define TIL     16

// ---- LDS layout (float offsets), padded strides for bank-conflict control --
#define QS_OFF   0                       // [16][65]    q tile
#define KS_OFF   (QS_OFF + 16*65)        // [16][65]    k tile
#define W1P_OFF  (KS_OFF + 16*65)        // [64 p][136] W1 prod-half^T (cols h)
#define W2T_OFF  (W1P_OFF + 64*136)      // [128 p][72] W2^T (cols n)
#define HB_OFF   (W2T_OFF + 128*72)      // [256 m][129] h buffer
#define AQ_OFF   (HB_OFF + 256*129)      // [16 j][136]  Aq tile
#define AK_OFF   (AQ_OFF + 16*136)       // [16 i][136]  Ak tile
#define B1_OFF   (AK_OFF + 16*136)       // [128]
#define LG_OFF   (B1_OFF + 128)          // [128]
#define LB_OFF   (LG_OFF + 128)          // [128]
#define B2_OFF   (LB_OFF + 128)          // [64]
#define SMEM_FLOATS (B2_OFF + 64)        // 57824 floats = 231296 bytes

// ---------------------------------------------------------------------------
// Kernel 1: fused downprojection + rank-terms
//   xd[l][p] = x[l,:].Wd[p,:] + bd[p]
//   aq[l][h] = sum_{p<64} xd[l][p]    * W1[h][64+p]
//   ak[l][h] = sum_{p<64} xd[l][64+p] * W1[h][64+p]
// ---------------------------------------------------------------------------
__global__ void pph_downproj(const float* __restrict__ x,
                             const float* __restrict__ Wd,
                             const float* __restrict__ bd,
                             const float* __restrict__ W1,
                             float* __restrict__ xd,
                             float* __restrict__ aq,
                             float* __restrict__ ak) {
    __shared__ float xrow[D_IN];
    __shared__ float qk[DP];
    const int l = blockIdx.x;
    for (int d = threadIdx.x; d < D_IN; d += blockDim.x)
        xrow[d] = x[l * D_IN + d];
    __syncthreads();
    const int p = threadIdx.x;          // 128 threads
    const float* w = Wd + p * D_IN;
    float acc = bd[p];
    #pragma unroll 8
    for (int d = 0; d < D_IN; ++d) acc = fmaf(xrow[d], w[d], acc);
    xd[l * DP + p] = acc;
    qk[p] = acc;
    __syncthreads();
    const float* w1d = W1 + p * DP + 64;    // diff-half row of W1 for h = p
    float aqv = 0.f, akv = 0.f;
    #pragma unroll 8
    for (int t = 0; t < 64; ++t) {
        float wv = w1d[t];
        aqv = fmaf(qk[t],      wv, aqv);
        akv = fmaf(qk[64 + t], wv, akv);
    }
    aq[l * DP + p] = aqv;
    ak[l * DP + p] = akv;
}

// ---------------------------------------------------------------------------
// Kernel 2: per 16x16 (i,j) tile. 256 threads = 8 wave32.
//   GEMM1 (K=64, prod features) -> +Aq-Ak+b1 -> GELU -> LN -> GEMM2 -> out
// ---------------------------------------------------------------------------
__global__ void
__launch_bounds__(256, 1)
pph_main(const float* __restrict__ xd,
         const float* __restrict__ aq, const float* __restrict__ ak,
         const float* __restrict__ W1, const float* __restrict__ b1,
         const float* __restrict__ ln_g, const float* __restrict__ ln_b,
         const float* __restrict__ W2, const float* __restrict__ b2,
         float* __restrict__ out) {
    extern __shared__ float smem[];
    float* qs  = smem + QS_OFF;
    float* ks  = smem + KS_OFF;
    float* w1p = smem + W1P_OFF;
    float* w2t = smem + W2T_OFF;
    float* hb  = smem + HB_OFF;
    float* aqs = smem + AQ_OFF;
    float* aks = smem + AK_OFF;
    float* b1s = smem + B1_OFF;
    float* lgs = smem + LG_OFF;
    float* lbs = smem + LB_OFF;
    float* b2s = smem + B2_OFF;

    const int tid  = threadIdx.x;
    const int lane = tid & 31;
    const int wid  = tid >> 5;            // 8 waves
    const int ln15 = lane & 15;
    const int hi   = lane >> 4;           // lane group 0/1
    const int hi2  = hi << 1;             // K offset inside wmma fragment
    const int hi8  = hi << 3;             // M offset of accumulator halves

    const int itile = blockIdx.y;         // k side
    const int jtile = blockIdx.x;         // q side

    // ---- stage tiles & weights into LDS --------------------------------
    #pragma unroll
    for (int it = 0; it < 4; ++it) {      // q,k 16x64 tiles
        int idx = tid + it * 256;         // 0..1023
        int r = idx >> 6, p = idx & 63;
        qs[r * 65 + p] = xd[(jtile * TIL + r) * DP + p];
        ks[r * 65 + p] = xd[(itile * TIL + r) * DP + 64 + p];
    }
    for (int it = 0; it < 32; ++it) {     // W1 prod half: [h][p<64] -> w1p[p][h]
        int idx = tid + it * 256;         // 0..8191
        int h = idx >> 6, p = idx & 63;
        w1p[p * 136 + h] = W1[h * DP + p];
    }
    for (int it = 0; it < 32; ++it) {     // W2 [n][p] -> w2t[p][n]
        int idx = tid + it * 256;         // 0..8191
        int n = idx >> 7, p = idx & 127;
        w2t[p * 72 + n] = W2[idx];
    }
    #pragma unroll
    for (int it = 0; it < 8; ++it) {      // Aq / Ak 16x128 tiles
        int idx = tid + it * 256;         // 0..2047
        int r = idx >> 7, h = idx & 127;
        aqs[r * 136 + h] = aq[(jtile * TIL + r) * DP + h];
        aks[r * 136 + h] = ak[(itile * TIL + r) * DP + h];
    }
    if (tid < 128) { b1s[tid] = b1[tid]; lgs[tid] = ln_g[tid]; lbs[tid] = ln_b[tid]; }
    if (tid < 64)  { b2s[tid] = b2[tid]; }
    __syncthreads();

    // ---- GEMM1: K=64 prod-feature GEMM. Each wave owns 2 M-tiles, loops
    //      8 N-tiles with the A fragments cached in registers. -------------
    for (int mm = 0; mm < 2; ++mm) {
        const int mt = wid * 2 + mm;                  // i_loc of this M-tile
        const float* qrow = qs + ln15 * 65;           // j_loc = ln15 (A rows)
        const float* krow = ks + mt * 65;             // broadcast row
        v2f acache[16];
        #pragma unroll
        for (int kk = 0; kk < 16; ++kk) {             // build prod A fragments
            int p = kk * 4 + hi2;
            acache[kk].x = qrow[p]     * krow[p];
            acache[kk].y = qrow[p + 1] * krow[p + 1];
        }
        for (int nt = 0; nt < 8; ++nt) {
            const int n = nt * 16 + ln15;
            v8f acc = {0.f, 0.f, 0.f, 0.f, 0.f, 0.f, 0.f, 0.f};
            #pragma unroll
            for (int kk = 0; kk < 16; ++kk) {
                int p = kk * 4 + hi2;
                v2f b; b.x = w1p[p * 136 + n]; b.y = w1p[(p + 1) * 136 + n];
                acc = __builtin_amdgcn_wmma_f32_16x16x4_f32(
                          false, acache[kk], false, b, (short)0, acc, false, false);
            }
            const float base = b1s[n] - aks[mt * 136 + n];   // -Ak[i] + b1
            #pragma unroll
            for (int r = 0; r < 8; ++r) {
                int jl = r + hi8;                     // j_loc of this row
                float v = acc[r] + base + aqs[jl * 136 + n]; // +Aq[j]
                v = 0.5f * v * (1.0f + erff(v * 0.70710678118654752f));
                hb[(mt * 16 + jl) * 129 + n] = v;
            }
        }
    }
    __syncthreads();

    // ---- LayerNorm over H=128, one thread per pair ----------------------
    {
        float* hr = hb + tid * 129;
        float s = 0.f;
        #pragma unroll 8
        for (int p = 0; p < 128; ++p) s += hr[p];
        const float mu = s * (1.0f / 128.0f);
        float var = 0.f;
        #pragma unroll 8
        for (int p = 0; p < 128; ++p) { float d = hr[p] - mu; var = fmaf(d, d, var); }
        const float inv = rsqrtf(var * (1.0f / 128.0f) + 1e-5f);
        #pragma unroll 8
        for (int p = 0; p < 128; ++p)
            hr[p] = (hr[p] - mu) * inv * lgs[p] + lbs[p];
    }
    __syncthreads();

    // ---- GEMM2: [256x128]x[128x64]. Wave owns 2 M-tiles, loops 4 N-tiles,
    //      A row fragments cached in registers. ---------------------------
    for (int mm = 0; mm < 2; ++mm) {
        const int mt = wid * 2 + mm;
        const float* arow = hb + (mt * 16 + ln15) * 129;
        v2f acache[32];
        #pragma unroll
        for (int kk = 0; kk < 32; ++kk) {
            int k0 = kk * 4 + hi2;
            acache[kk].x = arow[k0];
            acache[kk].y = arow[k0 + 1];
        }
        for (int nt = 0; nt < 4; ++nt) {
            const int n = nt * 16 + ln15;
            v8f acc = {0.f, 0.f, 0.f, 0.f, 0.f, 0.f, 0.f, 0.f};
            #pragma unroll
            for (int kk = 0; kk < 32; ++kk) {
                int k0 = kk * 4 + hi2;
                v2f b; b.x = w2t[k0 * 72 + n]; b.y = w2t[(k0 + 1) * 72 + n];
                acc = __builtin_amdgcn_wmma_f32_16x16x4_f32(
                          false, acache[kk], false, b, (short)0, acc, false, false);
            }
            const float bval = b2s[n];
            const int gi = itile * TIL + mt;
            #pragma unroll
            for (int r = 0; r < 8; ++r) {
                int gj = jtile * TIL + r + hi8;
                out[((size_t)gi * L_SEQ + gj) * NB + n] = acc[r] + bval;
            }
        }
    }
}

// ---------------------------------------------------------------------------
extern "C" void kernel_launch(void* const* d_in, const int* in_sizes, int n_in,
                              void* d_out, int out_size, void* d_ws, size_t ws_size,
                              hipStream_t stream) {
    const float* x    = (const float*)d_in[0];
    const float* Wd   = (const float*)d_in[1];
    const float* bd   = (const float*)d_in[2];
    const float* W1   = (const float*)d_in[3];
    const float* b1   = (const float*)d_in[4];
    const float* ln_g = (const float*)d_in[5];
    const float* ln_b = (const float*)d_in[6];
    const float* W2   = (const float*)d_in[7];
    const float* b2   = (const float*)d_in[8];
    float* out = (float*)d_out;

    float* xd = (float*)d_ws;                 // [768][128]
    float* aq = xd + L_SEQ * DP;              // [768][128]
    float* ak = aq + L_SEQ * DP;              // [768][128]  (~1.2 MB total)

    pph_downproj<<<L_SEQ, DP, 0, stream>>>(x, Wd, bd, W1, xd, aq, ak);

    dim3 grid(L_SEQ / TIL, L_SEQ / TIL);      // 48 x 48 tiles
    size_t smem = (size_t)SMEM_FLOATS * sizeof(float);   // ~226 KB < 320 KB/WGP
    pph_main<<<grid, 256, smem, stream>>>(xd, aq, ak, W1, b1, ln_g, ln_b,
                                          W2, b2, out);
}